// GPRPropagation_14645838480163
// MI455X (gfx1250) — compile-verified
//
#include <hip/hip_runtime.h>
#include <stdint.h>
#include <stddef.h>

#define F 128
#define THREADS 256
#define WAVES_PER_BLOCK (THREADS / 32)

// ---------------------------------------------------------------------------
// gfx1250 async global->LDS path (ASYNCcnt). Guarded so host pass / older
// toolchains still compile via a direct-load fallback.
// ---------------------------------------------------------------------------
#if defined(__gfx1250__) && __has_builtin(__builtin_amdgcn_global_load_async_to_lds_b128)
#define USE_ASYNC_LDS 1
#else
#define USE_ASYNC_LDS 0
#endif

#if USE_ASYNC_LDS
// Builtin signature (from clang diagnostic): first param is
// 'int __vector_size__(16) addrspace(1)*' (printed as '__device__'),
// second is the LDS-side pointer.
typedef int v4i_t __attribute__((vector_size(16)));
typedef __attribute__((address_space(1))) v4i_t glb_v4i_t;
typedef __attribute__((address_space(3))) v4i_t lds_v4i_t;
#endif

template <unsigned short N>
static __device__ __forceinline__ void wait_async() {
#if defined(__gfx1250__) && __has_builtin(__builtin_amdgcn_s_wait_asynccnt)
  __builtin_amdgcn_s_wait_asynccnt(N);
  asm volatile("" ::: "memory");  // keep LDS reads below the wait
#else
  asm volatile("s_wait_asynccnt %0" ::"i"((int)N) : "memory");
#endif
}

// ---------------------------------------------------------------------------
// Pass 1: in-degree of destination nodes (self loop added later as +1)
// ---------------------------------------------------------------------------
__global__ void count_deg_kernel(const int* __restrict__ ecol,
                                 unsigned int* __restrict__ deg, int E) {
  int e = blockIdx.x * blockDim.x + threadIdx.x;
  if (e < E) atomicAdd(&deg[ecol[e]], 1u);
}

// dis[i] = 1/sqrt(deg[i] + 1)
__global__ void make_dis_kernel(const unsigned int* __restrict__ deg,
                                float* __restrict__ dis, int N) {
  int i = blockIdx.x * blockDim.x + threadIdx.x;
  if (i < N) dis[i] = rsqrtf((float)deg[i] + 1.0f);
}

// norm[e] = dis[row[e]] * dis[col[e]]
__global__ void make_norm_kernel(const int* __restrict__ erow,
                                 const int* __restrict__ ecol,
                                 const float* __restrict__ dis,
                                 float* __restrict__ nrm, int E) {
  int e = blockIdx.x * blockDim.x + threadIdx.x;
  if (e < E) nrm[e] = dis[erow[e]] * dis[ecol[e]];
}

// out = w0 * x ; h1 = dis^2 * x   (self-loop contribution of round 1)
__global__ void init_out_h_kernel(const float* __restrict__ x,
                                  const float* __restrict__ dis,
                                  const float* __restrict__ wts,
                                  float* __restrict__ out,
                                  float* __restrict__ h, int total4) {
  int i = blockIdx.x * blockDim.x + threadIdx.x;
  if (i >= total4) return;
  const float w0 = wts[0];
  const int node = i >> 5;  // F/4 == 32 float4 chunks per row
  float s = dis[node];
  s *= s;
  float4 xv = ((const float4*)x)[i];
  float4 ov, hv;
  ov.x = w0 * xv.x; ov.y = w0 * xv.y; ov.z = w0 * xv.z; ov.w = w0 * xv.w;
  hv.x = s * xv.x;  hv.y = s * xv.y;  hv.z = s * xv.z;  hv.w = s * xv.w;
  ((float4*)out)[i] = ov;
  ((float4*)h)[i] = hv;
}

// out += w_k * h_cur ; (optionally) h_next = dis^2 * h_cur (self-loop of k+1)
__global__ void accum_init_kernel(const float* __restrict__ hcur,
                                  float* __restrict__ hnext,
                                  float* __restrict__ out,
                                  const float* __restrict__ wts, int k,
                                  const float* __restrict__ dis, int total4,
                                  int writeNext) {
  int i = blockIdx.x * blockDim.x + threadIdx.x;
  if (i >= total4) return;
  const float wk = wts[k];
  float4 hv = ((const float4*)hcur)[i];
  float4 ov = ((float4*)out)[i];
  ov.x += wk * hv.x; ov.y += wk * hv.y; ov.z += wk * hv.z; ov.w += wk * hv.w;
  ((float4*)out)[i] = ov;
  if (writeNext) {
    const int node = i >> 5;
    float s = dis[node];
    s *= s;
    float4 nv;
    nv.x = s * hv.x; nv.y = s * hv.y; nv.z = s * hv.z; nv.w = s * hv.w;
    ((float4*)hnext)[i] = nv;
  }
}

// ---------------------------------------------------------------------------
// Edge scatter: one wave (32 lanes) per edge, lane = float4 chunk of the row.
// Double-buffered async gather of h_old[row] into LDS (ASYNCcnt), then
// no-return global_atomic_add_f32 scatter into h_new[col]. Edge id is
// wave-uniform -> row/col/norm become scalar loads.
// ---------------------------------------------------------------------------
__global__ void __launch_bounds__(THREADS)
edge_propagate_kernel(const float* __restrict__ h_old,
                      float* __restrict__ h_new,
                      const int* __restrict__ erow,
                      const int* __restrict__ ecol,
                      const float* __restrict__ enorm, int E) {
#if USE_ASYNC_LDS
  __shared__ alignas(16) float smem[WAVES_PER_BLOCK][2][F];
#endif
  const int lane = threadIdx.x & 31;
  const int wave = threadIdx.x >> 5;
  const int stride = gridDim.x * WAVES_PER_BLOCK;
  int e = blockIdx.x * WAVES_PER_BLOCK + wave;
  if (e >= E) return;

#if USE_ASYNC_LDS
  {
    const int r = erow[e];
    const float* src = h_old + (size_t)r * F + lane * 4;
    __builtin_amdgcn_global_load_async_to_lds_b128(
        (glb_v4i_t*)src, (lds_v4i_t*)&smem[wave][0][lane * 4], 0, 0);
  }
  int buf = 0;
  for (;;) {
    const int e_next = e + stride;
    const bool more = (e_next < E);
    if (more) {
      const int r = erow[e_next];
      const float* src = h_old + (size_t)r * F + lane * 4;
      __builtin_amdgcn_global_load_async_to_lds_b128(
          (glb_v4i_t*)src, (lds_v4i_t*)&smem[wave][buf ^ 1][lane * 4], 0, 0);
      wait_async<1>();  // current buffer's gather has landed (in-order)
    } else {
      wait_async<0>();
    }
    const float nrm = enorm[e];
    const int c = ecol[e];
    const float4 v = *(const float4*)&smem[wave][buf][lane * 4];
    float* dst = h_new + (size_t)c * F + lane * 4;
    unsafeAtomicAdd(dst + 0, nrm * v.x);
    unsafeAtomicAdd(dst + 1, nrm * v.y);
    unsafeAtomicAdd(dst + 2, nrm * v.z);
    unsafeAtomicAdd(dst + 3, nrm * v.w);
    if (!more) break;
    e = e_next;
    buf ^= 1;
  }
#else
  for (; e < E; e += stride) {
    const int r = erow[e];
    const int c = ecol[e];
    const float nrm = enorm[e];
    const float4 v = *(const float4*)(h_old + (size_t)r * F + lane * 4);
    float* dst = h_new + (size_t)c * F + lane * 4;
    unsafeAtomicAdd(dst + 0, nrm * v.x);
    unsafeAtomicAdd(dst + 1, nrm * v.y);
    unsafeAtomicAdd(dst + 2, nrm * v.z);
    unsafeAtomicAdd(dst + 3, nrm * v.w);
  }
#endif
}

// ---------------------------------------------------------------------------
extern "C" void kernel_launch(void* const* d_in, const int* in_sizes, int n_in,
                              void* d_out, int out_size, void* d_ws,
                              size_t ws_size, hipStream_t stream) {
  (void)n_in; (void)out_size; (void)ws_size;
  const float* x = (const float*)d_in[0];
  const int* eidx = (const int*)d_in[1];   // [2, E] int32
  const float* wts = (const float*)d_in[2];

  const int NF = in_sizes[0];
  const int N = NF / F;
  const int E = in_sizes[1] / 2;
  const int order = in_sizes[2] - 1;  // 10

  const int* erow = eidx;       // source (gather)
  const int* ecol = eidx + E;   // destination (scatter)

  // workspace carve-out (256B aligned slices)
  uint8_t* ws = (uint8_t*)d_ws;
  size_t off = 0;
  auto carve = [&](size_t bytes) -> void* {
    void* p = (void*)(ws + off);
    off += (bytes + 255) & ~(size_t)255;
    return p;
  };
  unsigned int* deg = (unsigned int*)carve((size_t)N * 4);
  float* dis = (float*)carve((size_t)N * 4);
  float* nrm = (float*)carve((size_t)E * 4);
  float* hA = (float*)carve((size_t)NF * 4);
  float* hB = (float*)carve((size_t)NF * 4);

  (void)hipMemsetAsync(deg, 0, (size_t)N * 4, stream);

  const int tpb = THREADS;
  count_deg_kernel<<<(E + tpb - 1) / tpb, tpb, 0, stream>>>(ecol, deg, E);
  make_dis_kernel<<<(N + tpb - 1) / tpb, tpb, 0, stream>>>(deg, dis, N);
  make_norm_kernel<<<(E + tpb - 1) / tpb, tpb, 0, stream>>>(erow, ecol, dis, nrm, E);

  const int total4 = NF / 4;
  const int eg = (total4 + tpb - 1) / tpb;
  init_out_h_kernel<<<eg, tpb, 0, stream>>>(x, dis, wts, (float*)d_out, hA, total4);

  // ~2 edges per wave -> a 2-deep async pipeline per wave
  int waves_needed = (E + 1) / 2;
  int eblocks = (waves_needed + WAVES_PER_BLOCK - 1) / WAVES_PER_BLOCK;
  if (eblocks < 1) eblocks = 1;

  const float* hold = x;
  float* hnew = hA;
  for (int k = 1; k <= order; ++k) {
    edge_propagate_kernel<<<eblocks, tpb, 0, stream>>>(hold, hnew, erow, ecol, nrm, E);
    float* hnext = (hnew == hA) ? hB : hA;
    const int writeNext = (k < order) ? 1 : 0;
    accum_init_kernel<<<eg, tpb, 0, stream>>>(hnew, hnext, (float*)d_out, wts, k,
                                              dis, total4, writeNext);
    hold = hnew;
    hnew = hnext;
  }
}